// GCNModel_90091234000961
// MI455X (gfx1250) — compile-verified
//
#include <hip/hip_runtime.h>
#include <math.h>

#define NN 50000
#define NE 800000
#define DIM 128
#define NEG_SLOPE 0.01f

typedef float v2f __attribute__((ext_vector_type(2)));
typedef float v8f __attribute__((ext_vector_type(8)));

// ---------------- degree computation ----------------
__global__ void deg_kernel(const int* __restrict__ src, const int* __restrict__ dst,
                           float* __restrict__ deg_out, float* __restrict__ deg_in) {
    int i = blockIdx.x * blockDim.x + threadIdx.x;
    int stride = gridDim.x * blockDim.x;
    for (int e = i; e < NE; e += stride) {
        atomicAdd(&deg_out[src[e]], 1.0f);
        atomicAdd(&deg_in[dst[e]], 1.0f);
    }
}

// ---------------- feats = concat(weight, emb[sig]) * out_deg^-1/2 ----------------
__global__ void feats_kernel(const float* __restrict__ weight, const int* __restrict__ sig,
                             const float* __restrict__ emb, const float* __restrict__ deg_out,
                             float* __restrict__ h0) {
    int n = blockIdx.x * blockDim.x + threadIdx.x;
    if (n >= NN) return;
    float rs = rsqrtf(fmaxf(deg_out[n], 1.0f));
    int s = sig[n];
    h0[n * 3 + 0] = weight[n] * rs;
    h0[n * 3 + 1] = emb[s * 2 + 0] * rs;
    h0[n * 3 + 2] = emb[s * 2 + 1] * rs;
}

// ---------------- 3-wide edge scatter ----------------
__global__ void agg3_kernel(const int* __restrict__ src, const int* __restrict__ dst,
                            const float* __restrict__ h0, float* __restrict__ agg3) {
    int i = blockIdx.x * blockDim.x + threadIdx.x;
    int stride = gridDim.x * blockDim.x;
    for (int e = i; e < NE; e += stride) {
        int s = src[e] * 3, d = dst[e] * 3;
        atomicAdd(&agg3[d + 0], h0[s + 0]);
        atomicAdd(&agg3[d + 1], h0[s + 1]);
        atomicAdd(&agg3[d + 2], h0[s + 2]);
    }
}

// ---------------- layer0: in-scale, @W0+b0, LeakyReLU, out-scale (fused) ----------------
__global__ void layer0_kernel(const float* __restrict__ agg3, const float* __restrict__ deg_in,
                              const float* __restrict__ deg_out, const float* __restrict__ W0,
                              const float* __restrict__ b0, float* __restrict__ x1s) {
    int n = blockIdx.x;
    int f = threadIdx.x;
    float in_rs = rsqrtf(fmaxf(deg_in[n], 1.0f));
    float a0 = agg3[n * 3 + 0] * in_rs;
    float a1 = agg3[n * 3 + 1] * in_rs;
    float a2 = agg3[n * 3 + 2] * in_rs;
    float v = b0[f] + a0 * W0[f] + a1 * W0[DIM + f] + a2 * W0[2 * DIM + f];
    v = v > 0.0f ? v : NEG_SLOPE * v;
    float out_rs = rsqrtf(fmaxf(deg_out[n], 1.0f));
    x1s[(size_t)n * DIM + f] = v * out_rs;
}

// ---------------- 128-wide edge scatter: 32 lanes/edge, float4 per lane ----------------
__global__ void agg128_kernel(const int* __restrict__ src, const int* __restrict__ dst,
                              const float* __restrict__ x1s, float* __restrict__ agg) {
    long tid = (long)blockIdx.x * blockDim.x + threadIdx.x;
    long total = (long)NE * 32;
    long stride = (long)gridDim.x * blockDim.x;
    for (long t = tid; t < total; t += stride) {
        int e = (int)(t >> 5);
        int lane = (int)(t & 31);
        int s = src[e], d = dst[e];
        const float4 v = *(const float4*)&x1s[(size_t)s * DIM + lane * 4];
        float* p = &agg[(size_t)d * DIM + lane * 4];
        atomicAdd(p + 0, v.x);
        atomicAdd(p + 1, v.y);
        atomicAdd(p + 2, v.z);
        atomicAdd(p + 3, v.w);
    }
}

// ---------------- layer1 GEMM via V_WMMA_F32_16X16X4_F32 ----------------
// out[n,f] = (agg[n,:] * in_rs[n]) @ W1 + b1[f]
// one 16x16 output tile per wave; 8 waves/block cover all 128 columns;
// blockIdx.x covers the 3125 row tiles (50000 = 16*3125 exactly).
__global__ void gemm_wmma_kernel(const float* __restrict__ agg, const float* __restrict__ deg_in,
                                 const float* __restrict__ W1, const float* __restrict__ b1,
                                 float* __restrict__ out) {
    const int wave = threadIdx.x >> 5;   // column tile 0..7
    const int lane = threadIdx.x & 31;
    const int half = lane >> 4;          // 0: K pair {0,1}, 1: K pair {2,3}
    const int idx  = lane & 15;
    const int row  = blockIdx.x * 16 + idx;   // A row / lane's M index
    const int col  = wave * 16 + idx;         // B/C/D column (N index)

    const float rs = rsqrtf(fmaxf(deg_in[row], 1.0f));
    const float* __restrict__ arow = agg + (size_t)row * DIM;

    v8f c = {};
#pragma unroll
    for (int k = 0; k < DIM; k += 4) {
        const int ka = k + 2 * half;
        v2f a, b;
        a.x = arow[ka]     * rs;              // A[idx][ka]   (16x4 f32 layout)
        a.y = arow[ka + 1] * rs;              // A[idx][ka+1]
        b.x = W1[ka * DIM + col];             // B[ka][col]   (4x16 f32 layout)
        b.y = W1[(ka + 1) * DIM + col];       // B[ka+1][col]
        c = __builtin_amdgcn_wmma_f32_16x16x4_f32(
                /*neg_a=*/false, a, /*neg_b=*/false, b,
                /*c_mod=*/(short)0, c, /*reuse_a=*/false, /*reuse_b=*/false);
    }

    const float bias = b1[col];
#pragma unroll
    for (int j = 0; j < 8; ++j) {
        int r = blockIdx.x * 16 + 8 * half + j;   // D layout: VGPR j -> M=j (+8 for lanes 16-31)
        out[(size_t)r * DIM + col] = c[j] + bias;
    }
}

extern "C" void kernel_launch(void* const* d_in, const int* in_sizes, int n_in,
                              void* d_out, int out_size, void* d_ws, size_t ws_size,
                              hipStream_t stream) {
    (void)in_sizes; (void)n_in; (void)out_size; (void)ws_size;
    const int*   src    = (const int*)d_in[0];
    const int*   dst    = (const int*)d_in[1];
    const float* weight = (const float*)d_in[2];
    const int*   sig    = (const int*)d_in[3];
    const float* emb    = (const float*)d_in[4];
    const float* W0     = (const float*)d_in[5];
    const float* b0     = (const float*)d_in[6];
    const float* W1     = (const float*)d_in[7];
    const float* b1     = (const float*)d_in[8];
    float*       out    = (float*)d_out;

    // workspace layout (256B-aligned regions)
    char* ws = (char*)d_ws;
    size_t off = 0;
    auto carve = [&](size_t bytes) -> float* {
        float* p = (float*)(ws + off);
        off += (bytes + 255) & ~(size_t)255;
        return p;
    };
    float* deg_out = carve((size_t)NN * 4);
    float* deg_in  = carve((size_t)NN * 4);
    float* h0      = carve((size_t)NN * 3 * 4);
    float* agg3    = carve((size_t)NN * 3 * 4);
    float* x1s     = carve((size_t)NN * DIM * 4);
    float* agg     = carve((size_t)NN * DIM * 4);

    // deterministic re-init of accumulators every call (capture-safe)
    hipMemsetAsync(deg_out, 0, (size_t)NN * 4, stream);
    hipMemsetAsync(deg_in,  0, (size_t)NN * 4, stream);
    hipMemsetAsync(agg3,    0, (size_t)NN * 3 * 4, stream);
    hipMemsetAsync(agg,     0, (size_t)NN * DIM * 4, stream);

    deg_kernel<<<(NE + 255) / 256, 256, 0, stream>>>(src, dst, deg_out, deg_in);
    feats_kernel<<<(NN + 255) / 256, 256, 0, stream>>>(weight, sig, emb, deg_out, h0);
    agg3_kernel<<<(NE + 255) / 256, 256, 0, stream>>>(src, dst, h0, agg3);
    layer0_kernel<<<NN, DIM, 0, stream>>>(agg3, deg_in, deg_out, W0, b0, x1s);
    agg128_kernel<<<(NE * 32) / 256, 256, 0, stream>>>(src, dst, x1s, agg);
    gemm_wmma_kernel<<<NN / 16, 256, 0, stream>>>(agg, deg_in, W1, b1, out);
}